// GATLayer_5592047419814
// MI455X (gfx1250) — compile-verified
//
#include <hip/hip_runtime.h>
#include <hip/hip_bf16.h>

// GAT layer for MI455X (gfx1250, wave32).
// Pipeline: [zero-init] -> [WMMA f32 GEMM h=X@W^T] -> [per-node scores]
//        -> [edge pass1: leaky-relu + segment-max] -> [edge pass2: exp + segment-sum]
//        -> [weighted scatter-add aggregation] -> [in-place ELU].

typedef float v2f __attribute__((ext_vector_type(2)));
typedef float v8f __attribute__((ext_vector_type(8)));

#define IN_DIM  128
#define OUT_DIM 64
#define ALPHA_LEAKY 0.2f
#define EPS_GAT 1e-10f

// ---------------------------------------------------------------------------
// Kernel 1: zero-init of accumulators (d_out, e_sum, e_max keys)
// ---------------------------------------------------------------------------
__global__ void gat_zero_init(float* __restrict__ out, float* __restrict__ esum,
                              unsigned* __restrict__ emaxkey, int nOut, int nNodes) {
    int i = blockIdx.x * blockDim.x + threadIdx.x;
    if (i < nOut) out[i] = 0.0f;
    if (i < nNodes) { esum[i] = 0.0f; emaxkey[i] = 0u; } // key 0 == -inf sentinel
}

// ---------------------------------------------------------------------------
// Kernel 2: h = X @ W^T via V_WMMA_F32_16X16X4_F32 (wave32).
// One wave computes a 16(M) x 64(N) tile; K=128 in 32 steps of 4.
// W (64x128, 32KB) is staged in LDS; B-fragments come from ds_load_b64.
// ---------------------------------------------------------------------------
__global__ __launch_bounds__(256) void gat_gemm_wmma(
    const float* __restrict__ X, const float* __restrict__ W,
    float* __restrict__ h, int N) {
    __shared__ float Wlds[OUT_DIM * IN_DIM]; // 8192 floats = 32 KB

    const int tid = threadIdx.x;
    // Cooperative W load: 256 threads x 8 float4 = 8192 floats.
    for (int i = tid * 4; i < OUT_DIM * IN_DIM; i += 256 * 4) {
        *(float4*)&Wlds[i] = *(const float4*)&W[i];
    }
    __syncthreads();

    const int wave = tid >> 5;
    const int lane = tid & 31;
    const int m0 = (blockIdx.x * 8 + wave) * 16;
    if (m0 >= N) return; // wave-uniform exit; EXEC stays all-1s for WMMA

    const int nlo = lane & 15;       // N (or M for A) within half-wave
    const int khi = (lane >> 4) * 2; // K offset for upper half-wave

    v8f acc0 = {}, acc1 = {}, acc2 = {}, acc3 = {};

    const float* __restrict__ xrow = X + (size_t)(m0 + nlo) * IN_DIM + khi;
    const float* __restrict__ wrow = Wlds + nlo * IN_DIM + khi;

    #pragma unroll
    for (int k0 = 0; k0 < IN_DIM; k0 += 4) {
        // A-frag: lane L -> row m0+(L&15), K = k0 + 2*(L>>4) + {0,1}
        v2f a = *(const v2f*)(xrow + k0);
        // B-frag (B = W^T): lane L -> col nt*16+(L&15), same K split
        v2f b0 = *(const v2f*)(wrow + k0);
        v2f b1 = *(const v2f*)(wrow + 16 * IN_DIM + k0);
        v2f b2 = *(const v2f*)(wrow + 32 * IN_DIM + k0);
        v2f b3 = *(const v2f*)(wrow + 48 * IN_DIM + k0);
        acc0 = __builtin_amdgcn_wmma_f32_16x16x4_f32(false, a, false, b0, (short)0, acc0, false, false);
        acc1 = __builtin_amdgcn_wmma_f32_16x16x4_f32(false, a, false, b1, (short)0, acc1, false, false);
        acc2 = __builtin_amdgcn_wmma_f32_16x16x4_f32(false, a, false, b2, (short)0, acc2, false, false);
        acc3 = __builtin_amdgcn_wmma_f32_16x16x4_f32(false, a, false, b3, (short)0, acc3, false, false);
    }

    // C/D layout: VGPR v, lane L -> (M = v + 8*(L>>4), N = L&15)
    const int mbase = m0 + ((lane >> 4) << 3);
    #pragma unroll
    for (int v = 0; v < 8; ++v) {
        float* hr = h + (size_t)(mbase + v) * OUT_DIM + nlo;
        hr[0]  = acc0[v];
        hr[16] = acc1[v];
        hr[32] = acc2[v];
        hr[48] = acc3[v];
    }
}

// ---------------------------------------------------------------------------
// Kernel 3: per-node partial attention scores s_src = h.a[:64], s_tgt = h.a[64:]
// ---------------------------------------------------------------------------
__global__ void gat_scores(const float* __restrict__ h, const float* __restrict__ a,
                           float* __restrict__ s_src, float* __restrict__ s_tgt, int N) {
    int i = blockIdx.x * blockDim.x + threadIdx.x;
    if (i >= N) return;
    const float* hr = h + (size_t)i * OUT_DIM;
    float ss = 0.0f, st = 0.0f;
    #pragma unroll
    for (int d = 0; d < OUT_DIM; d += 4) {
        float4 hv = *(const float4*)(hr + d);
        float4 as = *(const float4*)(a + d);
        float4 at = *(const float4*)(a + OUT_DIM + d);
        ss += hv.x * as.x + hv.y * as.y + hv.z * as.z + hv.w * as.w;
        st += hv.x * at.x + hv.y * at.y + hv.z * at.z + hv.w * at.w;
    }
    s_src[i] = ss;
    s_tgt[i] = st;
}

// Order-preserving float->uint key: unsigned compare == float compare.
__device__ __forceinline__ unsigned f32_key(float x) {
    unsigned f = __float_as_uint(x);
    return (f & 0x80000000u) ? ~f : (f | 0x80000000u);
}
__device__ __forceinline__ float f32_unkey(unsigned k) {
    unsigned f = (k & 0x80000000u) ? (k ^ 0x80000000u) : ~k;
    return __uint_as_float(f);
}

// ---------------------------------------------------------------------------
// Kernel 4: e = leaky_relu(s_src[src]+s_tgt[tgt]); segment max over tgt.
// ---------------------------------------------------------------------------
__global__ void gat_edge_max(const int* __restrict__ src, const int* __restrict__ tgt,
                             const float* __restrict__ s_src, const float* __restrict__ s_tgt,
                             float* __restrict__ e_out, unsigned* __restrict__ emaxkey, int E) {
    int i = blockIdx.x * blockDim.x + threadIdx.x;
    if (i >= E) return;
    int s = src[i], t = tgt[i];
    float e = s_src[s] + s_tgt[t];
    e = (e > 0.0f) ? e : ALPHA_LEAKY * e;
    e_out[i] = e;
    atomicMax(&emaxkey[t], f32_key(e));
}

// ---------------------------------------------------------------------------
// Kernel 5: e_exp = exp(e - e_max[tgt]); segment sum over tgt.
// ---------------------------------------------------------------------------
__global__ void gat_edge_exp(const int* __restrict__ tgt, const float* __restrict__ e,
                             const unsigned* __restrict__ emaxkey,
                             float* __restrict__ ex, float* __restrict__ esum, int E) {
    int i = blockIdx.x * blockDim.x + threadIdx.x;
    if (i >= E) return;
    int t = tgt[i];
    float v = __expf(e[i] - f32_unkey(emaxkey[t]));
    ex[i] = v;
    atomicAdd(&esum[t], v);
}

// ---------------------------------------------------------------------------
// Kernel 6: h_prime[tgt] += h[src] * att. 64 threads (2 waves) per edge,
// so each wave is edge-uniform; gathers/atomics are 256B-coalesced.
// ---------------------------------------------------------------------------
__global__ void gat_aggregate(const int* __restrict__ src, const int* __restrict__ tgt,
                              const float* __restrict__ h, const float* __restrict__ ex,
                              const float* __restrict__ esum, float* __restrict__ out, int E) {
    long long gid = (long long)blockIdx.x * blockDim.x + threadIdx.x;
    int d = (int)(gid & (OUT_DIM - 1));
    long long eidx = gid >> 6;
    if (eidx >= E) return;
    int s = src[eidx], t = tgt[eidx];
    float att = ex[eidx] / (esum[t] + EPS_GAT);
    atomicAdd(&out[(size_t)t * OUT_DIM + d], h[(size_t)s * OUT_DIM + d] * att);
}

// ---------------------------------------------------------------------------
// Kernel 7: in-place ELU on accumulated h_prime.
// ---------------------------------------------------------------------------
__global__ void gat_elu(float* __restrict__ out, int n) {
    int i = blockIdx.x * blockDim.x + threadIdx.x;
    if (i >= n) return;
    float x = out[i];
    out[i] = (x > 0.0f) ? x : (__expf(x) - 1.0f);
}

// ---------------------------------------------------------------------------
extern "C" void kernel_launch(void* const* d_in, const int* in_sizes, int n_in,
                              void* d_out, int out_size, void* d_ws, size_t ws_size,
                              hipStream_t stream) {
    const float* X = (const float*)d_in[0];   // [N, 128]
    const int*   edge = (const int*)d_in[1];  // [2, E]
    const float* W = (const float*)d_in[2];   // [64, 128]
    const float* a = (const float*)d_in[3];   // [1, 128]
    float* out = (float*)d_out;               // [N, 64]

    const int N = in_sizes[0] / IN_DIM;
    const int E = in_sizes[1] / 2;
    const int* src = edge;
    const int* tgt = edge + E;

    // Workspace layout (floats)
    float*    h       = (float*)d_ws;          // N*64
    float*    e       = h + (size_t)N * OUT_DIM; // E
    float*    ex      = e + E;                 // E
    float*    ssrc    = ex + E;                // N
    float*    stgt    = ssrc + N;              // N
    float*    esum    = stgt + N;              // N
    unsigned* emaxkey = (unsigned*)(esum + N); // N

    const int nOut = N * OUT_DIM;

    // 1) zero accumulators (harness does not re-poison between replays)
    {
        int n = nOut > N ? nOut : N;
        gat_zero_init<<<(n + 255) / 256, 256, 0, stream>>>(out, esum, emaxkey, nOut, N);
    }
    // 2) h = X @ W^T  (WMMA f32 16x16x4, 8 waves/block, 16 rows/wave)
    {
        int mtiles = (N + 15) / 16;
        int blocks = (mtiles + 7) / 8;
        gat_gemm_wmma<<<blocks, 256, 0, stream>>>(X, W, h, N);
    }
    // 3) per-node scores
    gat_scores<<<(N + 255) / 256, 256, 0, stream>>>(h, a, ssrc, stgt, N);
    // 4) edge scores + segment max
    gat_edge_max<<<(E + 255) / 256, 256, 0, stream>>>(src, tgt, ssrc, stgt, e, emaxkey, E);
    // 5) exp + segment sum
    gat_edge_exp<<<(E + 255) / 256, 256, 0, stream>>>(tgt, e, emaxkey, ex, esum, E);
    // 6) weighted scatter-add aggregation (64 threads per edge)
    {
        long long threads = (long long)E * OUT_DIM;
        long long blocks = (threads + 255) / 256;
        gat_aggregate<<<(unsigned)blocks, 256, 0, stream>>>(src, tgt, h, ex, esum, out, E);
    }
    // 7) ELU in place
    gat_elu<<<(nOut + 255) / 256, 256, 0, stream>>>(out, nOut);
}